// TokensSamplerRegularizationLoss_20478404068049
// MI455X (gfx1250) — compile-verified
//
#include <hip/hip_runtime.h>
#include <stdint.h>

// Problem geometry from the reference setup (static): B x (16,160,160), sms=2
#define T_DIM 16
#define H_DIM 80
#define W_DIM 80
#define PLANE (H_DIM * W_DIM)   // 6400 floats = 25.6 KB
#define NPS   (T_DIM * PLANE)   // 102400 tokens per sample
#define BLOCK 256               // 8 wave32s
#define NWAVE (BLOCK / 32)

// ---- CDNA5 async global->LDS copy path (gfx1250) -------------------------
// Probe round 1 revealed the builtin prototype: param 0 is
// 'int __attribute__((vector_size(16))) *' (v4i) in the global AS.
typedef int v4i __attribute__((vector_size(4 * sizeof(int))));
typedef __attribute__((address_space(1))) v4i* g_v4i_ptr;
typedef __attribute__((address_space(3))) v4i* l_v4i_ptr;

#if defined(__gfx1250__) && __has_builtin(__builtin_amdgcn_global_load_async_to_lds_b128)
#define USE_ASYNC 1
__device__ __forceinline__ void async_cp16(const float* g, float* l) {
  __builtin_amdgcn_global_load_async_to_lds_b128(
      (g_v4i_ptr)(g),
      (l_v4i_ptr)(l),
      /*imm offset=*/0, /*cpol=*/0);
}
#if __has_builtin(__builtin_amdgcn_s_wait_asynccnt)
#define ASYNC_WAIT() __builtin_amdgcn_s_wait_asynccnt(0)
#else
#define ASYNC_WAIT() asm volatile("s_wait_asynccnt 0" ::: "memory")
#endif
#else
#define USE_ASYNC 0
#endif

// Full-wave butterfly sum (wave32): every lane ends with the wave total.
__device__ __forceinline__ float wave_sum32(float v) {
#pragma unroll
  for (int off = 16; off > 0; off >>= 1)
    v += __shfl_xor(v, off, 32);
  return v;
}

// One block per (sample b, time-plane z). Stages plane z (+ plane z+1 halo)
// into LDS, accumulates the 7 reduction partials, writes them to ws[bt*8..].
__global__ void __launch_bounds__(BLOCK)
plane_kernel(const float* __restrict__ scores, float* __restrict__ partial) {
  __shared__ __align__(16) float p0[PLANE];
  __shared__ __align__(16) float p1[PLANE];
  __shared__ float wsum[NWAVE][8];

  const int bt  = blockIdx.x;
  const int b   = bt / T_DIM;
  const int z   = bt % T_DIM;
  const int tid = threadIdx.x;
  const bool has_t = (z < T_DIM - 1);

  const float* src0 = scores + (size_t)b * NPS + (size_t)z * PLANE;

#if USE_ASYNC
  for (int i = tid; i < PLANE / 4; i += BLOCK)
    async_cp16(src0 + i * 4, &p0[i * 4]);
  if (has_t) {
    const float* src1 = src0 + PLANE;
    for (int i = tid; i < PLANE / 4; i += BLOCK)
      async_cp16(src1 + i * 4, &p1[i * 4]);
  }
  ASYNC_WAIT();        // my waves' async copies landed in LDS
  __syncthreads();     // everyone else's too
#else
  for (int i = tid; i < PLANE / 4; i += BLOCK)
    ((float4*)p0)[i] = ((const float4*)src0)[i];
  if (has_t) {
    const float4* src1 = (const float4*)(src0 + PLANE);
    for (int i = tid; i < PLANE / 4; i += BLOCK)
      ((float4*)p1)[i] = src1[i];
  }
  __syncthreads();
#endif

  const float sx = 2.0f / (float)(W_DIM - 1);
  const float sy = 2.0f / (float)(H_DIM - 1);

  float S = 0.f, Sx = 0.f, Sy = 0.f, Sr = 0.f;
  float Tw = 0.f, Th = 0.f, Tt = 0.f;

  for (int idx = tid; idx < PLANE; idx += BLOCK) {
    const int x = idx % W_DIM;
    const int y = idx / W_DIM;
    const float cv = fmaxf(p0[idx], 0.0f);
    const float xv = fmaf((float)x, sx, -1.0f);
    const float yv = fmaf((float)y, sy, -1.0f);
    S += cv;
    Sx = fmaf(cv, xv, Sx);
    Sy = fmaf(cv, yv, Sy);
    Sr = fmaf(cv, fmaf(xv, xv, yv * yv), Sr);
    if (x < W_DIM - 1) Tw += fabsf(fmaxf(p0[idx + 1], 0.0f) - cv);
    if (y < H_DIM - 1) Th += fabsf(fmaxf(p0[idx + W_DIM], 0.0f) - cv);
    if (has_t)         Tt += fabsf(fmaxf(p1[idx], 0.0f) - cv);
  }

  float acc[7] = {S, Sx, Sy, Sr, Tw, Th, Tt};
  const int lane = tid & 31;
  const int wid  = tid >> 5;
#pragma unroll
  for (int k = 0; k < 7; ++k) acc[k] = wave_sum32(acc[k]);
  if (lane == 0) {
#pragma unroll
    for (int k = 0; k < 7; ++k) wsum[wid][k] = acc[k];
  }
  __syncthreads();
  if (tid == 0) {
    float out[7] = {0.f, 0.f, 0.f, 0.f, 0.f, 0.f, 0.f};
    for (int w = 0; w < NWAVE; ++w)        // fixed order -> deterministic
      for (int k = 0; k < 7; ++k) out[k] += wsum[w][k];
    float* dst = partial + (size_t)bt * 8;
    for (int k = 0; k < 7; ++k) dst[k] = out[k];
    dst[7] = 0.f;
  }
}

// Single block: per-sample scalar math + deterministic mean over B.
__global__ void __launch_bounds__(BLOCK)
finalize_kernel(const float* __restrict__ partial, float* __restrict__ out, int B) {
  __shared__ float wsum[NWAVE];
  float local = 0.f;
  for (int b = threadIdx.x; b < B; b += BLOCK) {
    float S = 0.f, Sx = 0.f, Sy = 0.f, Sr = 0.f, Tw = 0.f, Th = 0.f, Tt = 0.f;
    for (int z = 0; z < T_DIM; ++z) {     // fixed z order
      const float* p = partial + ((size_t)b * T_DIM + z) * 8;
      S += p[0]; Sx += p[1]; Sy += p[2]; Sr += p[3];
      Tw += p[4]; Th += p[5]; Tt += p[6];
    }
    const float tot = fmaxf(S, 1e-6f);
    const float inv = 1.0f / tot;
    const float mux = Sx * inv, muy = Sy * inv;
    const float m2  = mux * mux + muy * muy;
    // sum p*((x-mux)^2+(y-muy)^2) = Sr/tot - 2*m2 + m2*(S/tot)
    const float compact = Sr * inv - 2.0f * m2 + m2 * (S * inv);
    const float tv =
        Th * (1.0f / (float)(T_DIM * (H_DIM - 1) * W_DIM)) +
        Tw * (1.0f / (float)(T_DIM * H_DIM * (W_DIM - 1))) +
        0.3f * Tt * (1.0f / (float)((T_DIM - 1) * H_DIM * W_DIM));
    local += compact + tv;
  }
  local = wave_sum32(local);
  const int lane = threadIdx.x & 31;
  const int wid  = threadIdx.x >> 5;
  if (lane == 0) wsum[wid] = local;
  __syncthreads();
  if (threadIdx.x == 0) {
    float s = 0.f;
    for (int w = 0; w < NWAVE; ++w) s += wsum[w];
    out[0] = s / (float)B;
  }
}

extern "C" void kernel_launch(void* const* d_in, const int* in_sizes, int n_in,
                              void* d_out, int out_size, void* d_ws, size_t ws_size,
                              hipStream_t stream) {
  const float* scores = (const float*)d_in[0];
  const int B = in_sizes[0] / NPS;          // 256 for the reference setup
  float* partial = (float*)d_ws;            // B*16*8 floats = 128 KB scratch

  plane_kernel<<<dim3((unsigned)(B * T_DIM)), dim3(BLOCK), 0, stream>>>(scores, partial);
  finalize_kernel<<<dim3(1), dim3(BLOCK), 0, stream>>>(partial, (float*)d_out, B);
}